// ALNN_VARIANT_4475355922758
// MI455X (gfx1250) — compile-verified
//
#include <hip/hip_runtime.h>
#include <hip/hip_bf16.h>
#include <stdint.h>

// Reference shapes: B=64, K=49, L=200, D=36. ref_time[k] = k * (48/(K-1)).
#define NB 64
#define NK 49
#define NL 200
#define ND 36
#define NJ 8              // l-rows per pipeline stage (= threadIdx.y extent)
#define NT (ND * NJ)      // 288 threads / block = 9 wave32
#define NSTAGE (NL / NJ)  // 25 stages
#define BB 4              // batch elements register-tiled per block (weight reuse)

// ---- CDNA5 async global->LDS path (ASYNCcnt-tracked DMA) for the weight streams ----
__device__ __forceinline__ unsigned lds_off(const void* p) {
  // LDS aperture keeps the byte offset in addr[31:0].
  return (unsigned)(uintptr_t)p;
}
__device__ __forceinline__ void async_b32(unsigned loff, const void* g) {
  asm volatile("global_load_async_to_lds_b32 %0, %1, off"
               :: "v"(loff), "v"(g) : "memory");
}
__device__ __forceinline__ void async_b128(unsigned loff, const void* g) {
  asm volatile("global_load_async_to_lds_b128 %0, %1, off"
               :: "v"(loff), "v"(g) : "memory");
}
#define WAIT_ASYNC(n) asm volatile("s_wait_asynccnt " #n ::: "memory")

__global__ __launch_bounds__(NT) void alnn_fused_kernel(
    const float* __restrict__ X,  const float* __restrict__ T,
    const float* __restrict__ M,  const float* __restrict__ PD,
    const float* __restrict__ alpha, const float* __restrict__ w_v,
    const float* __restrict__ w_t,   const float* __restrict__ b_v,
    const float* __restrict__ b_t,   float* __restrict__ out)
{
  // Double-buffered async staging for the per-(k,l,d) weight streams only;
  // each thread's weight element is reused for all BB batches (4x traffic cut).
  __shared__ float4 sWT[2][NT];     // w_t[k,l,d,0:4] as one b128 per element
  __shared__ float  sBT[2][NT];
  __shared__ float  sWV[2][NT];
  __shared__ float  red[NJ][BB][ND];

  const int d  = threadIdx.x;           // 0..35  (variable)
  const int j  = threadIdx.y;           // 0..7   (l phase)
  const int t  = j * ND + d;            // 0..287
  const int k  = blockIdx.x;            // 0..48  (ref time point)
  const int b0 = blockIdx.y * BB;       // batch tile base

  const int wtbase = k  * (NL * ND) + t;   // (k*L + l)*D + d, l = s*8+j -> + s*NT
  const int xbase  = b0 * (NL * ND) + t;   // X/T/M/PD[b0, l, d]        -> + s*NT

  auto issueW = [&](int p, int s) {
    const int wi = wtbase + s * NT;
    async_b128(lds_off(&sWT[p][t]), w_t + (size_t)wi * 4);  // 16B aligned
    async_b32 (lds_off(&sBT[p][t]), b_t + wi);
    async_b32 (lds_off(&sWV[p][t]), w_v + wi);
  };

  // Per-block scalars: relu(alpha[k]) pre-scaled by log2(e) for v_exp_f32.
  const float a   = fmaxf(alpha[k], 0.0f);
  const float nal = -a * 1.4426950408889634f;
  const float kf  = (float)k * (48.0f / (float)(NK - 1));  // linspace(0,48,49)

  float acc[BB];
#pragma unroll
  for (int bb = 0; bb < BB; ++bb) acc[bb] = 0.0f;

  issueW(0, 0);

  for (int s = 0; s < NSTAGE; ++s) {
    const int p = s & 1;
    if (s + 1 < NSTAGE) { issueW(p ^ 1, s + 1); WAIT_ASYNC(3); }  // retire stage s
    else                { WAIT_ASYNC(0); }

    // Direct coalesced loads for the per-batch data (self-consumed: registers
    // beat an LDS round-trip). 16 independent loads -> deep MLP per wave.
    const int base = xbase + s * NT;
    float xv[BB], tv[BB], mv[BB], pv[BB];
#pragma unroll
    for (int bb = 0; bb < BB; ++bb) {
      const int idx = base + bb * (NL * ND);
      xv[bb] = X[idx];  tv[bb] = T[idx];
      mv[bb] = M[idx];  pv[bb] = PD[idx];
    }

    const float4 wt  = sWT[p][t];
    const float  bt4 = 4.0f * sBT[p][t];   // b_t broadcast over 4 channels
    const float  wv  = sWV[p][t];

#pragma unroll
    for (int bb = 0; bb < BB; ++bb) {
      const float dist  = fabsf(tv[bb] - kf);
      const float kern  = __builtin_amdgcn_exp2f(nal * dist);  // exp(-relu(a)*dist)
      const float inten = fmaxf(xv[bb] * kern, 0.0f);
      float s1 = fmaf(wt.x, xv[bb], bt4);
      s1       = fmaf(wt.y, inten,  s1);
      s1       = fmaf(wt.z, mv[bb], s1);
      s1       = fmaf(wt.w, pv[bb], s1);
      s1       = fmaxf(s1, 0.0f);
      acc[bb]  = fmaf(s1, wv, acc[bb]);     // stage-2 partial over this j's l's
    }
  }

  // Reduce the 8 l-phase partials per (bb, d); add L*b_v; relu; store [B,K,D].
#pragma unroll
  for (int bb = 0; bb < BB; ++bb) red[j][bb][d] = acc[bb];
  __syncthreads();

  if (j == 0) {
    const float bias = (float)NL * b_v[k * ND + d];
#pragma unroll
    for (int bb = 0; bb < BB; ++bb) {
      float sum = red[0][bb][d];
#pragma unroll
      for (int q = 1; q < NJ; ++q) sum += red[q][bb][d];
      out[((size_t)(b0 + bb) * NK + k) * ND + d] = fmaxf(sum + bias, 0.0f);
    }
  }
}

extern "C" void kernel_launch(void* const* d_in, const int* in_sizes, int n_in,
                              void* d_out, int out_size, void* d_ws, size_t ws_size,
                              hipStream_t stream) {
  (void)in_sizes; (void)n_in; (void)out_size; (void)d_ws; (void)ws_size;
  const float* X     = (const float*)d_in[0];
  const float* T     = (const float*)d_in[1];
  const float* M     = (const float*)d_in[2];
  const float* PD    = (const float*)d_in[3];
  const float* alpha = (const float*)d_in[4];
  const float* w_v   = (const float*)d_in[5];
  const float* w_t   = (const float*)d_in[6];
  const float* b_v   = (const float*)d_in[7];
  const float* b_t   = (const float*)d_in[8];
  float* out = (float*)d_out;

  dim3 grid(NK, NB / BB);   // 49 x 16 = 784 blocks
  dim3 block(ND, NJ);       // 36 x 8  = 288 threads (9 wave32)
  alnn_fused_kernel<<<grid, block, 0, stream>>>(X, T, M, PD, alpha, w_v, w_t,
                                                b_v, b_t, out);
}